// HR_HLIFLayer2D_42382737277004
// MI455X (gfx1250) — compile-verified
//
#include <hip/hip_runtime.h>
#include <math.h>

// Problem shape (fixed by the reference)
#define B_ 8
#define T_ 64
#define C_ 64
#define H_ 32
#define W_ 32

constexpr unsigned CHW  = C_ * H_ * W_;   // 65536 elements per (b,t) slice
constexpr unsigned QCHW = CHW / 4;        // 16384 float4-quads per slice

typedef float v4f __attribute__((ext_vector_type(4)));

__device__ __forceinline__ float softplus_f(float z) {
    // jax.nn.softplus: max(z,0) + log1p(exp(-|z|))  (overflow-safe form)
    return fmaxf(z, 0.0f) + log1pf(expf(-fabsf(z)));
}

__device__ __forceinline__ float sigmoid_f(float z) {
    return 1.0f / (1.0f + expf(-z));
}

__global__ __launch_bounds__(256) void HR_HLIF2D_fwd(
    const v4f* __restrict__ x,          // [B,T,QCHW]
    const v4f* __restrict__ vth_raw,    // [QCHW]
    const v4f* __restrict__ decay_raw,  // [QCHW]
    const float* __restrict__ hp_v_m,
    const float* __restrict__ hp_v_s,
    const float* __restrict__ hp_d_m,
    const float* __restrict__ hp_d_s,
    v4f* __restrict__ out)              // [B,T,QCHW]
{
    // blockIdx.y = batch (uniform -> scalar address math);
    // blockIdx.x * 256 + tid = quad index within the [C,H,W] slice.
    const unsigned e = blockIdx.x * blockDim.x + threadIdx.x;  // < QCHW
    const unsigned b = blockIdx.y;                             // < B_

    // Uniform hyper-parameter scalars (1-element device arrays -> s_load)
    const float v_m = *hp_v_m;
    const float v_s = *hp_v_s;
    const float d_m = *hp_d_m;
    const float d_s = *hp_d_s;

    // Per-element threshold & decay (512KB raw params -> L2 resident)
    const v4f vr = vth_raw[e];
    const v4f dr = decay_raw[e];

    v4f vth, dec;
    #pragma unroll
    for (int c = 0; c < 4; ++c) {
        vth[c] = softplus_f(vr[c] * v_s + v_m) + 0.5f;
        float dd = sigmoid_f(dr[c] * d_s + d_m);
        dec[c] = fminf(fmaxf(dd, 0.0f), 0.99f);
    }

    // Base pointers for this (b, quad); time stride = QCHW quads (256KB)
    const size_t base = (size_t)b * (T_ * QCHW) + e;
    const v4f* __restrict__ xb = x   + base;
    v4f* __restrict__       ob = out + base;

    v4f v = (v4f)0.0f;

    // T is the serial dimension; the x_t loads are independent of the
    // recurrence, so process in chunks of 8 and software-prefetch the next
    // chunk (gfx1250 global_prefetch_b8) to hide the 256KB time stride.
    #pragma unroll 1
    for (unsigned t0 = 0; t0 < T_; t0 += 8) {
        if (t0 + 8 < T_) {
            #pragma unroll
            for (unsigned k = 0; k < 8; ++k)
                __builtin_prefetch((const void*)(xb + (size_t)(t0 + 8 + k) * QCHW), 0, 0);
        }
        #pragma unroll
        for (unsigned k = 0; k < 8; ++k) {
            const size_t off = (size_t)(t0 + k) * QCHW;
            // streaming read: touched exactly once -> non-temporal
            const v4f xv = __builtin_nontemporal_load(xb + off);
            v4f s;
            #pragma unroll
            for (int c = 0; c < 4; ++c) {
                v[c] = fmaf(v[c], dec[c], xv[c]);             // leaky integrate
                s[c] = (v[c] - vth[c] > 0.0f) ? 1.0f : 0.0f;  // heaviside spike
                v[c] = fmaf(-s[c], vth[c], v[c]);             // soft reset
            }
            // streaming write: never re-read -> non-temporal
            __builtin_nontemporal_store(s, ob + off);
        }
    }
}

extern "C" void kernel_launch(void* const* d_in, const int* in_sizes, int n_in,
                              void* d_out, int out_size, void* d_ws, size_t ws_size,
                              hipStream_t stream) {
    // setup_inputs order: x, vth_raw, decay_raw, hp_v_m, hp_v_s, hp_d_m, hp_d_s, hp_alpha
    const v4f*   x         = (const v4f*)d_in[0];
    const v4f*   vth_raw   = (const v4f*)d_in[1];
    const v4f*   decay_raw = (const v4f*)d_in[2];
    const float* hp_v_m    = (const float*)d_in[3];
    const float* hp_v_s    = (const float*)d_in[4];
    const float* hp_d_m    = (const float*)d_in[5];
    const float* hp_d_s    = (const float*)d_in[6];
    // d_in[7] = hp_alpha: surrogate-gradient sharpness, unused in forward

    v4f* out = (v4f*)d_out;

    dim3 block(256, 1, 1);
    dim3 grid(QCHW / 256, B_, 1);   // (64, 8) blocks, 8 wave32s per block
    HR_HLIF2D_fwd<<<grid, block, 0, stream>>>(
        x, vth_raw, decay_raw, hp_v_m, hp_v_s, hp_d_m, hp_d_s, out);
}